// Attention_85847806312552
// MI455X (gfx1250) — compile-verified
//
#include <hip/hip_runtime.h>
#include <stdint.h>

typedef float v2f __attribute__((ext_vector_type(2)));
typedef float v8f __attribute__((ext_vector_type(8)));

#define B_SZ 512
#define N_SZ 196
#define E_SZ 2048
#define D_SZ 512
#define A_SZ 512

// main GEMM tiling
#define KC        16          // K chunk staged in LDS (double buffered)
#define NCHUNK    (E_SZ / KC) // 128
#define MT        13          // 13 M-tiles of 16 rows (208 >= 196)
#define MROWS     (MT * 16)   // 208
#define EO_STRIDE 20          // padded LDS row stride (floats) -> kills bank conflicts
#define ABLK      128         // a-columns per workgroup (8 waves x 16)

// ---------------------------------------------------------------- helpers
// CDNA5 async global->LDS copy (16B per lane), tracked by ASYNCcnt.
__device__ __forceinline__ void async_ld128(unsigned lds_off, const float* g) {
    asm volatile("global_load_async_to_lds_b128 %0, %1, off"
                 :: "v"(lds_off), "v"(g)
                 : "memory");
}
__device__ __forceinline__ void wait_async_le6() {
    asm volatile("s_wait_asynccnt 0x6" ::: "memory");
}
__device__ __forceinline__ void wait_async_0() {
    asm volatile("s_wait_asynccnt 0x0" ::: "memory");
}

// ---------------------------------------------------------------- K0: zero
__global__ void zero_kernel(float* __restrict__ p, int n) {
    int i = blockIdx.x * 256 + threadIdx.x;
    if (i < n) p[i] = 0.f;
}

// ------------------------------------------- K1: att2 = dec@W_dec + b_dec + b_enc
__global__ __launch_bounds__(128)
void att2_kernel(const float* __restrict__ dec, const float* __restrict__ W_dec,
                 const float* __restrict__ b_dec, const float* __restrict__ b_enc,
                 float* __restrict__ att2) {
    const int b = blockIdx.x, tid = threadIdx.x;
    __shared__ float dh[D_SZ];
    for (int i = tid; i < D_SZ; i += 128) dh[i] = dec[b * D_SZ + i];
    __syncthreads();
    float a0 = b_dec[tid      ] + b_enc[tid      ];
    float a1 = b_dec[tid + 128] + b_enc[tid + 128];
    float a2 = b_dec[tid + 256] + b_enc[tid + 256];
    float a3 = b_dec[tid + 384] + b_enc[tid + 384];
    for (int k = 0; k < D_SZ; ++k) {
        const float d = dh[k];
        const float* wr = W_dec + (size_t)k * A_SZ + tid;
        a0 = fmaf(d, wr[0],   a0);
        a1 = fmaf(d, wr[128], a1);
        a2 = fmaf(d, wr[256], a2);
        a3 = fmaf(d, wr[384], a3);
    }
    att2[b * A_SZ + tid      ] = a0;
    att2[b * A_SZ + tid + 128] = a1;
    att2[b * A_SZ + tid + 256] = a2;
    att2[b * A_SZ + tid + 384] = a3;
}

// ---- K2: per-(b, a-block) WMMA GEMM att1 = Eo@W_enc, fused relu/W_full epilogue
//      Double-buffered LDS pipeline fed by global_load_async_to_lds_b128.
__global__ __launch_bounds__(256)
void gemm_score_kernel(const float* __restrict__ Eo, const float* __restrict__ W_enc,
                       const float* __restrict__ att2, const float* __restrict__ W_full,
                       float* __restrict__ score) {
    const int b    = blockIdx.y;
    const int ab0  = blockIdx.x * ABLK;
    const int tid  = threadIdx.x;
    const int lane = tid & 31;
    const int wv   = tid >> 5;      // wave id 0..7 -> N-tile
    const int half = lane >> 4;     // 0: lanes 0-15, 1: lanes 16-31
    const int l16  = lane & 15;

    __shared__ float EoT[2][MROWS * EO_STRIDE];   // 2 x 208 x 20 floats
    __shared__ float Wt[2][KC * ABLK];            // 2 x 16 x 128 floats
    __shared__ float dump[64];                    // sink for masked async lanes

    const float* Eo_b = Eo + (size_t)b * N_SZ * E_SZ;

    // prologue: zero pad rows 196-199 and 204-207 (cols 0..15) of both Eo buffers; async
    // loads never write these rows, so they stay zero for the whole kernel.
    for (int i = tid; i < 12 * 16 * 2; i += 256) {
        int bufi = i >= 12 * 16;
        int j    = i & (12 * 16 - 1);
        int r    = N_SZ + (j >> 4);
        EoT[bufi][r * EO_STRIDE + (j & 15)] = 0.f;
    }

    v8f acc[MT];
    #pragma unroll
    for (int m = 0; m < MT; ++m)
        #pragma unroll
        for (int j = 0; j < 8; ++j) acc[m][j] = 0.f;

    // ---- async stage of one K-chunk: every thread issues exactly 6 loads
    // (uniform per wave => deterministic ASYNCcnt), invalid units -> dump.
    auto stage = [&](int buf, int k0) {
        float* eoB = &EoT[buf][0];
        float* wtB = &Wt[buf][0];
        unsigned dumpOff = (unsigned)(uintptr_t)&dump[(tid & 15) * 4];
        #pragma unroll
        for (int t = 0; t < 4; ++t) {                 // 784 units padded to 1024
            int  i     = tid + t * 256;
            int  r     = i >> 2;
            int  c4    = (i & 3) * 4;
            bool valid = i < N_SZ * 4;                // rows 0..195 only
            const float* g  = valid ? (Eo_b + (size_t)r * E_SZ + k0 + c4) : Eo_b;
            unsigned     ld = valid ? (unsigned)(uintptr_t)(eoB + r * EO_STRIDE + c4)
                                    : dumpOff;
            async_ld128(ld, g);
        }
        #pragma unroll
        for (int t = 0; t < 2; ++t) {                 // 512 units, exact
            int i  = tid + t * 256;
            int r  = i >> 5;
            int c4 = (i & 31) * 4;
            async_ld128((unsigned)(uintptr_t)(wtB + r * ABLK + c4),
                        W_enc + (size_t)(k0 + r) * A_SZ + ab0 + c4);
        }
    };

    stage(0, 0);

    for (int c = 0; c < NCHUNK; ++c) {
        const int cur = c & 1;
        if (c + 1 < NCHUNK) {
            stage(cur ^ 1, (c + 1) * KC);   // overlap next chunk's traffic
            wait_async_le6();               // this wave's chunk-c loads done
        } else {
            wait_async_0();
        }
        __syncthreads();                    // all waves' chunk-c data visible

        const float* eoB = &EoT[cur][0];
        const float* wtB = &Wt[cur][0];
        #pragma unroll
        for (int kk = 0; kk < KC; kk += 4) {
            const int kA = kk + half * 2;   // K-pair owned by this lane half
            v2f bf;                          // B frag: VGPR0={K0|K2}, VGPR1={K1|K3}
            bf.x = wtB[(kA + 0) * ABLK + wv * 16 + l16];
            bf.y = wtB[(kA + 1) * ABLK + wv * 16 + l16];
            #pragma unroll
            for (int m = 0; m < MT; ++m) {
                v2f af = *(const v2f*)&eoB[(m * 16 + l16) * EO_STRIDE + kA];
                acc[m] = __builtin_amdgcn_wmma_f32_16x16x4_f32(
                    false, af, false, bf, (short)0, acc[m], false, false);
            }
        }
        __syncthreads();                    // buffer 'cur' free for reuse
    }

    // epilogue: relu(att1 + att2) * W_full, reduce the wave's 16 a-columns,
    // atomically accumulate partial scores (sum of relu terms splits over a-blocks)
    const int   aCol  = ab0 + wv * 16 + l16;
    const float biasv = att2[b * A_SZ + aCol];
    const float wfv   = W_full[aCol];
    #pragma unroll
    for (int m = 0; m < MT; ++m) {
        #pragma unroll
        for (int j = 0; j < 8; ++j) {
            float v = fmaxf(acc[m][j] + biasv, 0.f) * wfv;
            v += __shfl_xor(v, 1, 32);
            v += __shfl_xor(v, 2, 32);
            v += __shfl_xor(v, 4, 32);
            v += __shfl_xor(v, 8, 32);
            const int row = m * 16 + j + half * 8;   // C layout: VGPR j -> rows j / j+8
            if (l16 == 0 && row < N_SZ)
                atomicAdd(&score[b * N_SZ + row], v);
        }
    }
}

// ---------------------------------------------------------- K3: softmax over N
__global__ __launch_bounds__(256)
void softmax_kernel(float* __restrict__ alpha, const float* __restrict__ b_full) {
    const int b = blockIdx.x, tid = threadIdx.x;
    __shared__ float red[256];
    float x = -3.4e38f;
    if (tid < N_SZ) x = alpha[b * N_SZ + tid] + b_full[0];
    red[tid] = x;
    __syncthreads();
    for (int s = 128; s > 0; s >>= 1) {
        if (tid < s) red[tid] = fmaxf(red[tid], red[tid + s]);
        __syncthreads();
    }
    const float mx = red[0];
    __syncthreads();
    float e = (tid < N_SZ) ? __expf(x - mx) : 0.f;
    red[tid] = e;
    __syncthreads();
    for (int s = 128; s > 0; s >>= 1) {
        if (tid < s) red[tid] += red[tid + s];
        __syncthreads();
    }
    const float inv = 1.0f / red[0];
    if (tid < N_SZ) alpha[b * N_SZ + tid] = e * inv;
}

// ------------------------------------------------- K4: awe[b,e] = sum_n a*Eo
__global__ __launch_bounds__(256)
void awe_kernel(const float* __restrict__ Eo, const float* __restrict__ alpha,
                float* __restrict__ awe) {
    const int b  = blockIdx.y;
    const int e0 = blockIdx.x * 1024 + threadIdx.x * 4;
    __shared__ float al[N_SZ];
    if (threadIdx.x < N_SZ) al[threadIdx.x] = alpha[b * N_SZ + threadIdx.x];
    __syncthreads();
    const float* base = Eo + (size_t)b * N_SZ * E_SZ + e0;
    float4 s = make_float4(0.f, 0.f, 0.f, 0.f);
    for (int n = 0; n < N_SZ; ++n) {
        if (n + 8 < N_SZ) __builtin_prefetch(base + (size_t)(n + 8) * E_SZ, 0, 0);
        float4 v = *(const float4*)(base + (size_t)n * E_SZ);
        const float a = al[n];
        s.x = fmaf(a, v.x, s.x);
        s.y = fmaf(a, v.y, s.y);
        s.z = fmaf(a, v.z, s.z);
        s.w = fmaf(a, v.w, s.w);
    }
    *(float4*)&awe[(size_t)b * E_SZ + e0] = s;
}

// ----------------------------------------------------------------- launcher
extern "C" void kernel_launch(void* const* d_in, const int* in_sizes, int n_in,
                              void* d_out, int out_size, void* d_ws, size_t ws_size,
                              hipStream_t stream) {
    const float* Eo     = (const float*)d_in[0];   // [512,196,2048]
    const float* dec    = (const float*)d_in[1];   // [512,512]
    const float* W_enc  = (const float*)d_in[2];   // [2048,512]
    const float* b_enc  = (const float*)d_in[3];   // [512]
    const float* W_dec  = (const float*)d_in[4];   // [512,512]
    const float* b_dec  = (const float*)d_in[5];   // [512]
    const float* W_full = (const float*)d_in[6];   // [512,1]
    const float* b_full = (const float*)d_in[7];   // [1]

    float* awe   = (float*)d_out;                  // [512,2048]
    float* alpha = awe + (size_t)B_SZ * E_SZ;      // [512,196] (also score scratch)
    float* att2  = (float*)d_ws;                   // [512,512] f32 = 1 MB scratch

    const int score_n = B_SZ * N_SZ;
    zero_kernel<<<dim3((score_n + 255) / 256), 256, 0, stream>>>(alpha, score_n);
    att2_kernel<<<dim3(B_SZ), 128, 0, stream>>>(dec, W_dec, b_dec, b_enc, att2);
    gemm_score_kernel<<<dim3(A_SZ / ABLK, B_SZ), 256, 0, stream>>>(Eo, W_enc, att2, W_full, alpha);
    softmax_kernel<<<dim3(B_SZ), 256, 0, stream>>>(alpha, b_full);
    awe_kernel<<<dim3(E_SZ / 1024, B_SZ), 256, 0, stream>>>(Eo, alpha, awe);
}